// VectorQuantizerEMA_82970178224789
// MI455X (gfx1250) — compile-verified
//
#include <hip/hip_runtime.h>
#include <hip/hip_bf16.h>

#define N_TOK 65536
#define DIM   128
#define KCODE 1024
#define DECAY 0.99f
#define EPS   1e-05f

typedef __attribute__((ext_vector_type(2))) float v2f;
typedef __attribute__((ext_vector_type(8))) float v8f;

// ---------------------------------------------------------------------------
// Kernel 0: enorm[k] = sum(E[k,:]^2)
// ---------------------------------------------------------------------------
__global__ __launch_bounds__(256) void vq_enorm_kernel(const float* __restrict__ E,
                                                       float* __restrict__ enorm) {
    int k = blockIdx.x * blockDim.x + threadIdx.x;
    if (k < KCODE) {
        const float4* row = (const float4*)(E + (size_t)k * DIM);
        float s = 0.f;
#pragma unroll
        for (int i = 0; i < DIM / 4; ++i) {
            float4 v = row[i];
            s += v.x * v.x + v.y * v.y + v.z * v.z + v.w * v.w;
        }
        enorm[k] = s;
    }
}

// ---------------------------------------------------------------------------
// Kernel 1: distances via WMMA f32 16x16x4, argmin, gather, loss, scatter.
// Block = 256 threads (8 waves); each wave owns 32 tokens (two 16x16 C tiles
// sharing every B fragment -> 4 WMMAs per ds_load_2addr_b64, two independent
// accumulation chains). E tiles double-buffered in LDS with prefetch.
// ---------------------------------------------------------------------------
__global__ __launch_bounds__(256, 1) void vq_assign_kernel(
    const float* __restrict__ X, const float* __restrict__ E,
    const float* __restrict__ enorm,
    float* __restrict__ outQuant, float* __restrict__ outIdx,
    float* __restrict__ dwCount, float* __restrict__ dwEmbed,
    float* __restrict__ lossSum) {
    __shared__ float lsE[2][16 * 132];  // double-buffered E tile, pad 4 (bank-conflict free)
    __shared__ float lsEn[KCODE];       // |e|^2 staged per block
    __shared__ int   lsIdx[8][32];      // per-wave winning code per token

    const int tid  = threadIdx.x;
    const int wave = tid >> 5;
    const int lane = tid & 31;
    const int l16  = lane & 15;
    const int hi   = lane >> 4;       // 0 for lanes 0-15, 1 for 16-31
    const int coff = hi * 2;          // K-offset within 4-wide WMMA step

    const int waveTokBase = blockIdx.x * 256 + wave * 32;

    // ---- stage |e|^2 into LDS (covered by first barrier) ----
#pragma unroll
    for (int i = 0; i < KCODE / 256; ++i) lsEn[tid + i * 256] = enorm[tid + i * 256];

    // ---- preload A fragments: wave's 2 x (16x128) X tiles, ISA 16x4 f32 A layout ----
    const float* xrow0 = X + (size_t)(waveTokBase + l16) * DIM + coff;
    const float* xrow1 = xrow0 + 16 * DIM;
    v2f a0[32], a1[32];
#pragma unroll
    for (int d = 0; d < 32; ++d) {
        a0[d] = *(const v2f*)(xrow0 + d * 4);
        a1[d] = *(const v2f*)(xrow1 + d * 4);
    }

    // ---- stage E tile 0 (each thread moves 8 floats) ----
    const int sr = tid >> 4;          // 0..15 row of tile
    const int sc = (tid & 15) * 8;    // col start
    const float* esrc = E + (size_t)sr * DIM + sc;
    float4 p0 = *(const float4*)(esrc);
    float4 p1 = *(const float4*)(esrc + 4);
    {
        float* dst = &lsE[0][sr * 132 + sc];
        *(float4*)(dst)     = p0;
        *(float4*)(dst + 4) = p1;
    }
    __syncthreads();

    float minD0[8], minD1[8];
    int   minI0[8], minI1[8];
#pragma unroll
    for (int r = 0; r < 8; ++r) {
        minD0[r] = 3.4e38f; minI0[r] = 0;
        minD1[r] = 3.4e38f; minI1[r] = 0;
    }

    for (int kt = 0; kt < KCODE / 16; ++kt) {
        const int cur = kt & 1;
        const int cb  = kt * 16;
        // prefetch next tile into registers (overlaps with WMMA burst)
        if (kt + 1 < KCODE / 16) {
            const float* s = esrc + (size_t)(kt + 1) * 16 * DIM;
            p0 = *(const float4*)(s);
            p1 = *(const float4*)(s + 4);
        }

        v8f acc0 = {};
        v8f acc1 = {};
        const float* brow = &lsE[cur][l16 * 132 + coff];
#pragma unroll
        for (int d = 0; d < 32; ++d) {
            v2f b = *(const v2f*)(brow + d * 4);
            acc0 = __builtin_amdgcn_wmma_f32_16x16x4_f32(
                false, a0[d], false, b, (short)0, acc0, false, false);
            acc1 = __builtin_amdgcn_wmma_f32_16x16x4_f32(
                false, a1[d], false, b, (short)0, acc1, false, false);
        }

        const float en = lsEn[cb + l16];
#pragma unroll
        for (int r = 0; r < 8; ++r) {
            float d0 = en - 2.0f * acc0[r];   // + |x|^2 constant, irrelevant to argmin
            float d1 = en - 2.0f * acc1[r];
            if (d0 < minD0[r]) { minD0[r] = d0; minI0[r] = cb + l16; }
            if (d1 < minD1[r]) { minD1[r] = d1; minI1[r] = cb + l16; }
        }

        // write prefetched tile into the other buffer; one barrier per tile
        if (kt + 1 < KCODE / 16) {
            float* dst = &lsE[cur ^ 1][sr * 132 + sc];
            *(float4*)(dst)     = p0;
            *(float4*)(dst + 4) = p1;
        }
        __syncthreads();
    }

    // ---- cross-lane argmin within each 16-lane N-group (first-min tiebreak) ----
#pragma unroll
    for (int r = 0; r < 8; ++r) {
#pragma unroll
        for (int s = 1; s < 16; s <<= 1) {
            float od0 = __shfl_xor(minD0[r], s, 32);
            int   oi0 = __shfl_xor(minI0[r], s, 32);
            if (od0 < minD0[r] || (od0 == minD0[r] && oi0 < minI0[r])) {
                minD0[r] = od0; minI0[r] = oi0;
            }
            float od1 = __shfl_xor(minD1[r], s, 32);
            int   oi1 = __shfl_xor(minI1[r], s, 32);
            if (od1 < minD1[r] || (od1 == minD1[r] && oi1 < minI1[r])) {
                minD1[r] = od1; minI1[r] = oi1;
            }
        }
        if (l16 == 0) {
            lsIdx[wave][r + hi * 8]      = minI0[r];
            lsIdx[wave][16 + r + hi * 8] = minI1[r];
        }
    }
    __syncthreads();

    // ---- gather E[idx], write quantized, loss partial, scatter segment-sum ----
    float lsum = 0.f;
    const int c4 = lane * 4;
    for (int t = 0; t < 32; ++t) {
        const int tok = waveTokBase + t;
        const int k   = lsIdx[wave][t];
        float4 q = *(const float4*)(E + (size_t)k * DIM + c4);
        float4 x = *(const float4*)(X + (size_t)tok * DIM + c4);
        *(float4*)(outQuant + (size_t)tok * DIM + c4) = q;
        float d0 = q.x - x.x, d1 = q.y - x.y, d2 = q.z - x.z, d3 = q.w - x.w;
        lsum += d0 * d0 + d1 * d1 + d2 * d2 + d3 * d3;
        float* de = dwEmbed + (size_t)k * DIM + c4;
        atomicAdd(de + 0, x.x);
        atomicAdd(de + 1, x.y);
        atomicAdd(de + 2, x.z);
        atomicAdd(de + 3, x.w);
        if (lane == 0) {
            atomicAdd(&dwCount[k], 1.0f);
            outIdx[tok] = (float)k;
        }
    }
#pragma unroll
    for (int s = 1; s < 32; s <<= 1) lsum += __shfl_xor(lsum, s, 32);
    if (lane == 0) atomicAdd(lossSum, lsum);
}

// ---------------------------------------------------------------------------
// Kernel 2: new_cluster_size EMA + n = sum(new_cluster_size) + loss finalize
// ---------------------------------------------------------------------------
__global__ __launch_bounds__(1024) void vq_cluster_kernel(
    const float* __restrict__ clusterSize, const float* __restrict__ dwCount,
    float* __restrict__ outNCS, float* __restrict__ nSum,
    const float* __restrict__ lossSum, float* __restrict__ outLoss) {
    __shared__ float red[32];
    const int t = threadIdx.x;
    float ncs = clusterSize[t] * DECAY + (1.f - DECAY) * dwCount[t];
    outNCS[t] = ncs;
    float s = ncs;
#pragma unroll
    for (int sh = 1; sh < 32; sh <<= 1) s += __shfl_xor(s, sh, 32);
    if ((t & 31) == 0) red[t >> 5] = s;
    __syncthreads();
    if (t < 32) {
        float v = red[t];
#pragma unroll
        for (int sh = 1; sh < 32; sh <<= 1) v += __shfl_xor(v, sh, 32);
        if (t == 0) {
            *nSum = v;
            // loss = diff2 + COMMITMENT_COST * diff2 = 2 * mean((q - x)^2)
            *outLoss = 2.0f * (*lossSum) / (float)((size_t)N_TOK * DIM);
        }
    }
}

// ---------------------------------------------------------------------------
// Kernel 3: new_embed_avg EMA and new_embedding = new_embed_avg / cs
// ---------------------------------------------------------------------------
__global__ __launch_bounds__(128) void vq_embed_kernel(
    const float* __restrict__ embedAvg, const float* __restrict__ dwEmbed,
    const float* __restrict__ outNCS, const float* __restrict__ nSum,
    float* __restrict__ outNEA, float* __restrict__ outNewEmb) {
    const int k = blockIdx.x;
    const int c = threadIdx.x;
    const size_t o = (size_t)k * DIM + c;
    float nea = embedAvg[o] * DECAY + (1.f - DECAY) * dwEmbed[o];
    outNEA[o] = nea;
    float n   = *nSum;
    float ncs = outNCS[k];
    float cs  = (ncs + EPS) / (n + (float)KCODE * EPS) * n;
    outNewEmb[o] = nea / cs;
}

// ---------------------------------------------------------------------------
extern "C" void kernel_launch(void* const* d_in, const int* in_sizes, int n_in,
                              void* d_out, int out_size, void* d_ws, size_t ws_size,
                              hipStream_t stream) {
    const float* X           = (const float*)d_in[0];  // [65536,128]
    const float* E           = (const float*)d_in[1];  // [1024,128]
    const float* clusterSize = (const float*)d_in[2];  // [1024]
    const float* embedAvg    = (const float*)d_in[3];  // [1024,128]

    float* out = (float*)d_out;
    float* outQuant  = out;                                   // [65536,128]
    float* outLoss   = outQuant + (size_t)N_TOK * DIM;        // [1]
    float* outIdx    = outLoss + 1;                           // [65536] (as float)
    float* outNewEmb = outIdx + N_TOK;                        // [1024,128]
    float* outNCS    = outNewEmb + (size_t)KCODE * DIM;       // [1024]
    float* outNEA    = outNCS + KCODE;                        // [1024,128]

    float* ws       = (float*)d_ws;
    float* lossSum  = ws + 0;
    float* nSum     = ws + 1;
    float* enorm    = ws + 16;
    float* dwCount  = ws + 16 + KCODE;
    float* dwEmbed  = ws + 16 + 2 * KCODE;
    const size_t wsFloats = 16 + 2 * KCODE + (size_t)KCODE * DIM;

    hipMemsetAsync(d_ws, 0, wsFloats * sizeof(float), stream);
    vq_enorm_kernel<<<KCODE / 256, 256, 0, stream>>>(E, enorm);
    vq_assign_kernel<<<N_TOK / 256, 256, 0, stream>>>(
        X, E, enorm, outQuant, outIdx, dwCount, dwEmbed, lossSum);
    vq_cluster_kernel<<<1, 1024, 0, stream>>>(clusterSize, dwCount, outNCS, nSum,
                                              lossSum, outLoss);
    vq_embed_kernel<<<KCODE, DIM, 0, stream>>>(embedAvg, dwEmbed, outNCS, nSum,
                                               outNEA, outNewEmb);
}